// MultiResHashEncoding_13889924235701
// MI455X (gfx1250) — compile-verified
//
#include <hip/hip_runtime.h>
#include <math.h>
#include <stdint.h>

#define L_LEVELS 16
#define T_SIZE   16384
#define F_FEAT   2
#define CHUNK    8192
#define BLOCK    256
#define TBL_BYTES (T_SIZE * F_FEAT * 4)   // 131072 bytes = 128 KB per level

struct LevelScales { float s[L_LEVELS]; };

__global__ __launch_bounds__(BLOCK)
void mrhash_kernel(const float* __restrict__ pts,
                   const float* __restrict__ tables,
                   float* __restrict__ out,
                   LevelScales lv, int npts)
{
    extern __shared__ char smem[];            // dynamic LDS: 128 KB table for this level
    const int level = blockIdx.y;
    const int tid   = threadIdx.x;

    // ---- Stage this level's 128KB hash table into LDS with CDNA5 async
    //      global->LDS copies (ASYNCcnt path), 16B per lane per issue.
    {
        const char* gtab = (const char*)tables + (size_t)level * TBL_BYTES;
        unsigned lds_base = (unsigned)(uintptr_t)smem;   // low 32 bits of generic = LDS offset
        for (int ofs = tid * 16; ofs < TBL_BYTES; ofs += BLOCK * 16) {
            unsigned l = lds_base + (unsigned)ofs;
            unsigned g = (unsigned)ofs;
            asm volatile("global_load_async_to_lds_b128 %0, %1, %2"
                         :: "v"(l), "v"(g), "s"(gtab)
                         : "memory");
        }
        asm volatile("s_wait_asynccnt 0" ::: "memory");
    }
    __syncthreads();

    const float2* __restrict__ tbl = (const float2*)smem;
    const float scale = lv.s[level];
    const int base = blockIdx.x * CHUNK;

    for (int i = tid; i < CHUNK; i += BLOCK) {
        const int b = base + i;
        if (b >= npts) break;

        const float px = pts[(size_t)b * 3 + 0];
        const float py = pts[(size_t)b * 3 + 1];
        const float pz = pts[(size_t)b * 3 + 2];

        const float sx = scale * px, sy = scale * py, sz = scale * pz;
        const float fx = floorf(sx), fy = floorf(sy), fz = floorf(sz);
        const float ex = ceilf(sx) - fx;   // 0.0 or 1.0
        const float ey = ceilf(sy) - fy;
        const float ez = ceilf(sz) - fz;

        float a0 = 0.0f, a1 = 0.0f;

        #pragma unroll
        for (int k = 0; k < 8; ++k) {
            // IDXS row k = [(k>>2)&1, (k>>1)&1, k&1]
            const float nx = fx + (((k >> 2) & 1) ? ex : 0.0f);
            const float ny = fy + (((k >> 1) & 1) ? ey : 0.0f);
            const float nz = fz + (((k     ) & 1) ? ez : 0.0f);

            const int ix = (int)nx;   // non-negative, integer-valued
            const int iy = (int)ny;
            const int iz = (int)nz;

            // distance of int-cast grid corner to the UNSCALED point (faithful to ref)
            const float ddx = (float)ix - px;
            const float ddy = (float)iy - py;
            const float ddz = (float)iz - pz;
            const float dist = sqrtf(ddx * ddx + ddy * ddy + ddz * ddz);

            // int64 hash (1, 2654435761, 805459861) XOR, mod 2^14
            // == uint32 arithmetic on the low bits
            const unsigned h = ((unsigned)ix)
                             ^ ((unsigned)iy * 2654435761u)
                             ^ ((unsigned)iz * 805459861u);
            const float2 f = tbl[h & (unsigned)(T_SIZE - 1)];   // ds_load_b64 gather

            a0 = fmaf(f.x, dist, a0);
            a1 = fmaf(f.y, dist, a1);
        }

        // out[b, level*2 + {0,1}]
        float2* o = (float2*)(out + (size_t)b * (L_LEVELS * F_FEAT) + level * F_FEAT);
        *o = make_float2(a0, a1);
    }
}

extern "C" void kernel_launch(void* const* d_in, const int* in_sizes, int n_in,
                              void* d_out, int out_size, void* d_ws, size_t ws_size,
                              hipStream_t stream)
{
    const float* pts    = (const float*)d_in[0];   // [BS,3] f32
    const float* tables = (const float*)d_in[1];   // [L,T,F] f32
    float* out          = (float*)d_out;           // [BS, L*F] f32

    const int npts = in_sizes[0] / 3;

    // levels = NMIN * growth^l computed in double, cast to f32 (matches numpy)
    LevelScales lv;
    const double growth = exp((log(512.0) - log(16.0)) / 15.0);
    for (int l = 0; l < L_LEVELS; ++l)
        lv.s[l] = (float)(16.0 * pow(growth, (double)l));

    dim3 grid((npts + CHUNK - 1) / CHUNK, L_LEVELS);
    hipLaunchKernelGGL(mrhash_kernel, grid, dim3(BLOCK), TBL_BYTES, stream,
                       pts, tables, out, lv, npts);
}